// SpikingDenseLayer_11381663334461
// MI455X (gfx1250) — compile-verified
//
#include <hip/hip_runtime.h>

// Problem constants (from reference setup_inputs)
#define B_DIM 128
#define T_DIM 100
#define I_DIM 1024
#define O_DIM 1024
#define LDIM  1024      // leading dim of every matrix here
#define EPSV  1e-8f

typedef __attribute__((ext_vector_type(2))) float v2f;
typedef __attribute__((ext_vector_type(8))) float v8f;

// ---------------------------------------------------------------------------
// WMMA fragment mapping (V_WMMA_F32_16X16X4_F32, wave32):
//   A (16x4):  lanes 0-15 -> M=lane, V0=K0,V1=K1 ; lanes 16-31 -> M=lane-16, V0=K2,V1=K3
//   B (4x16):  V0: lanes0-15 = B[K0][N=lane], lanes16-31 = B[K2][N=lane-16]
//              V1: lanes0-15 = B[K1][N],      lanes16-31 = B[K3][N]
//   C/D(16x16):VGPR g: lanes0-15 = C[M=g][N=lane], lanes16-31 = C[M=g+8][N=lane-16]
// ---------------------------------------------------------------------------
#define WMMA_F32(a, b, c) \
  __builtin_amdgcn_wmma_f32_16x16x4_f32(false, (a), false, (b), (short)0, (c), false, false)

// Async global->LDS copy, 16B per lane, GV mode (saddr = off). Tracked by ASYNCcnt.
__device__ __forceinline__ void async_copy_b128(unsigned int lds_byte_off, const void* g) {
  asm volatile("global_load_async_to_lds_b128 %0, %1, off"
               :: "v"(lds_byte_off), "v"((unsigned long long)g)
               : "memory");
}
template <int N>
__device__ __forceinline__ void wait_async() {
  asm volatile("s_wait_asynccnt %0" :: "n"(N) : "memory");
}

// Generic pointers to __shared__ carry the wave-relative LDS offset in the
// low 32 bits (AS3 offset).
__device__ __forceinline__ unsigned int lds_off32(const void* p) {
  return (unsigned int)(unsigned long long)(uintptr_t)p;
}

// ---------------------------------------------------------------------------
// (4*MW) x 64 output-tile GEMM core, 128 threads = 4 waves, MW rows per wave
// (MW = 16 or 32; MW=32 -> two A-fragments share the four B-fragments, i.e.
// 1.5 LDS loads per WMMA instead of 2.5). K staged in 16-wide chunks with
// DOUBLE-BUFFERED async global->LDS copies (ASYNCcnt pipeline).
//   ATRANS=false : acc += A[m0+..][k] * B[k][n0+..]        (A row-major)
//   ATRANS=true  : acc += A[k][m0+..] * B[k][n0+..]        (A^T, e.g. w^T w)
// ---------------------------------------------------------------------------
template <bool ATRANS, int MW>
__device__ __forceinline__ void gemm_core(const float* __restrict__ A,
                                          const float* __restrict__ B,
                                          int K, int m0, int n0,
                                          v8f (&acc)[MW / 16][4]) {
  constexpr int BROWS  = 4 * MW;                    // block rows: 64 or 128
  constexpr int ASTR   = ATRANS ? (BROWS + 4) : 20; // padded, mult of 4 -> 16B aligned
  constexpr int AELEMS = ATRANS ? 16 * ASTR : BROWS * 20;
  constexpr int BELEMS = 16 * 68;
  constexpr int ACOPY  = (BROWS * 16) / (4 * 128);  // per-thread B128s for the A tile
  constexpr int AF4    = BROWS / 4;                 // float4s per k-row (ATRANS layout)

  __shared__ float As[2][AELEMS];
  __shared__ float Bs[2][BELEMS];

  const int tid  = threadIdx.x;
  const int lane = tid & 31;
  const int wv   = tid >> 5;     // 0..3 -> MW-row stripe
  const int half = lane >> 4;    // 0/1
  const int lrow = lane & 15;
  const unsigned int aBase = lds_off32(&As[0][0]);
  const unsigned int bBase = lds_off32(&Bs[0][0]);

  auto issue = [&](int buf, int k0) {
    // A tile
#pragma unroll
    for (int i = 0; i < ACOPY; ++i) {
      const int idx = tid + i * 128;
      if (ATRANS) {
        const int r = idx / AF4, c4 = (idx % AF4) * 4;   // r = k row, c4 = m offset
        async_copy_b128(aBase + (unsigned int)(buf * AELEMS + r * ASTR + c4) * 4u,
                        &A[(size_t)(k0 + r) * LDIM + m0 + c4]);
      } else {
        const int r = idx >> 2, c4 = (idx & 3) * 4;      // r = m row, c4 = k offset
        async_copy_b128(aBase + (unsigned int)(buf * AELEMS + r * 20 + c4) * 4u,
                        &A[(size_t)(m0 + r) * LDIM + k0 + c4]);
      }
    }
    // B tile (2 x B128 per thread)
#pragma unroll
    for (int i = 0; i < 2; ++i) {
      const int idx = tid + i * 128;
      const int r = idx >> 4, c4 = (idx & 15) * 4;
      async_copy_b128(bBase + (unsigned int)(buf * BELEMS + r * 68 + c4) * 4u,
                      &B[(size_t)(k0 + r) * LDIM + n0 + c4]);
    }
  };

  issue(0, 0);                       // prologue: prefetch chunk 0
  int cur = 0;
  for (int k0 = 0; k0 < K; k0 += 16) {
    const bool more = (k0 + 16) < K;
    if (more) {
      issue(cur ^ 1, k0 + 16);       // overlap next chunk's copy with compute
      wait_async<ACOPY + 2>();       // newest batch in flight -> current chunk landed
    } else {
      wait_async<0>();
    }
    __syncthreads();                 // all waves' tiles visible

#pragma unroll
    for (int ks = 0; ks < 16; ks += 4) {
      v2f a[MW / 16];
#pragma unroll
      for (int j = 0; j < MW / 16; ++j) {
        if (ATRANS) {
          a[j].x = As[cur][(ks + 2 * half)     * ASTR + MW * wv + 16 * j + lrow];
          a[j].y = As[cur][(ks + 2 * half + 1) * ASTR + MW * wv + 16 * j + lrow];
        } else {
          a[j].x = As[cur][(MW * wv + 16 * j + lrow) * 20 + ks + 2 * half];
          a[j].y = As[cur][(MW * wv + 16 * j + lrow) * 20 + ks + 2 * half + 1];
        }
      }
      const float* bp0 = &Bs[cur][(ks + 2 * half)     * 68];
      const float* bp1 = &Bs[cur][(ks + 2 * half + 1) * 68];
      v2f b[4];
#pragma unroll
      for (int nt = 0; nt < 4; ++nt) {
        b[nt].x = bp0[16 * nt + lrow];
        b[nt].y = bp1[16 * nt + lrow];
      }
#pragma unroll
      for (int j = 0; j < MW / 16; ++j)
#pragma unroll
        for (int nt = 0; nt < 4; ++nt)
          acc[j][nt] = WMMA_F32(a[j], b[nt], acc[j][nt]);
    }
    __syncthreads();                 // compute done before buffer reuse
    cur ^= 1;
  }
}

// ================= h = x @ w  (M=12800, N=K=1024), 128x64 tiles =============
__global__ __launch_bounds__(128)
void gemm_h_kernel(const float* __restrict__ x, const float* __restrict__ w,
                   float* __restrict__ h) {
  const int m0 = blockIdx.y * 128;
  const int n0 = blockIdx.x * 64;
  const int tid = threadIdx.x, lane = tid & 31, wv = tid >> 5;
  const int half = lane >> 4, lrow = lane & 15;

  v8f acc[2][4] = {};
  gemm_core<false, 32>(x, w, I_DIM, m0, n0, acc);

#pragma unroll
  for (int j = 0; j < 2; ++j)
#pragma unroll
    for (int g = 0; g < 8; ++g) {
      const size_t row = (size_t)(m0 + 32 * wv + 16 * j + g + 8 * half);
#pragma unroll
      for (int nt = 0; nt < 4; ++nt)
        h[row * O_DIM + n0 + 16 * nt + lrow] = acc[j][nt][g];
    }
}

// ==== E = (1-beta)*v - beta*(w^T w); inv_norm = 1/(diag+eps); 128x64 tiles ===
__global__ __launch_bounds__(128)
void gemm_e_kernel(const float* __restrict__ w, const float* __restrict__ vmat,
                   const float* __restrict__ beta,
                   float* __restrict__ E, float* __restrict__ inv_norm) {
  const int m0 = blockIdx.y * 128;
  const int n0 = blockIdx.x * 64;
  const int tid = threadIdx.x, lane = tid & 31, wv = tid >> 5;
  const int half = lane >> 4, lrow = lane & 15;

  v8f acc[2][4] = {};
  gemm_core<true, 32>(w, w, I_DIM, m0, n0, acc);          // acc = (w^T w) tile

  const float bet = beta[0];
  const float omb = 1.0f - bet;
#pragma unroll
  for (int j = 0; j < 2; ++j)
#pragma unroll
    for (int g = 0; g < 8; ++g) {
      const int m = m0 + 32 * wv + 16 * j + g + 8 * half;
      const size_t mrow = (size_t)m * O_DIM;
#pragma unroll
      for (int nt = 0; nt < 4; ++nt) {
        const int n = n0 + 16 * nt + lrow;
        const float d = acc[j][nt][g];
        E[mrow + n] = omb * vmat[mrow + n] - bet * d;
        if (m == n) inv_norm[n] = 1.0f / (d + EPSV);      // norm[o] = (w^T w)[o][o]
      }
    }
}

// ======================= one scan step (t fixed) ============================
// u = spk_in @ E ; mem = beta*mem + (1-beta)*h_t + u ; spk = (mem*inv_norm - b > 0)
// h_t is read from out[:,t,:] and overwritten in place with spk.
// Kept at 64x64 tiles (32 blocks) to maximize WGP spread on the latency-
// critical sequential part.
__global__ __launch_bounds__(128)
void scan_step_kernel(const float* __restrict__ spk_in, const float* __restrict__ E,
                      const float* __restrict__ beta, const float* __restrict__ bvec,
                      const float* __restrict__ inv_norm,
                      float* __restrict__ mem, float* __restrict__ spk_out,
                      float* __restrict__ out, float* __restrict__ tsum, int t) {
  __shared__ float red[128];
  const int m0 = blockIdx.y * 64;   // batch rows  (grid.y = 2)
  const int n0 = blockIdx.x * 64;   // neurons     (grid.x = 16)
  const int tid = threadIdx.x, lane = tid & 31, wv = tid >> 5;
  const int half = lane >> 4, lrow = lane & 15;

  v8f acc[1][4] = {};
  gemm_core<false, 16>(spk_in, E, O_DIM, m0, n0, acc);

  const float bet = beta[0];
  const float omb = 1.0f - bet;
  float cnt = 0.0f;
#pragma unroll
  for (int g = 0; g < 8; ++g) {
    const int b = m0 + 16 * wv + g + 8 * half;    // batch index
    const size_t memrow = (size_t)b * O_DIM;
    const size_t hrow   = ((size_t)b * T_DIM + t) * (size_t)O_DIM;
#pragma unroll
    for (int nt = 0; nt < 4; ++nt) {
      const int o = n0 + 16 * nt + lrow;
      const float u = acc[0][nt][g];
      const float m_new = bet * mem[memrow + o] + omb * out[hrow + o] + u;
      const float mthr  = m_new * inv_norm[o] - bvec[o];
      const float s     = (mthr > 0.0f) ? 1.0f : 0.0f;
      mem[memrow + o]     = m_new;
      spk_out[memrow + o] = s;
      out[hrow + o]       = s;      // overwrite h with spike record
      cnt += s;
    }
  }
  red[tid] = cnt;
  __syncthreads();
  if (tid == 0) {
    float ssum = 0.0f;
#pragma unroll 4
    for (int i = 0; i < 128; ++i) ssum += red[i];
    atomicAdd(&tsum[t], ssum);
  }
}

// ================================ misc ======================================
__global__ void init_zero_kernel(float* __restrict__ p, int n) {
  int i = blockIdx.x * blockDim.x + threadIdx.x;
  if (i < n) p[i] = 0.0f;
}

__global__ void finalize_kernel(const float* __restrict__ tsum, float* __restrict__ out) {
  if (threadIdx.x == 0) {
    float s = 0.0f, mx = 0.0f;
    for (int t = 0; t < T_DIM; ++t) {
      const float v = tsum[t];
      s += v;
      mx = fmaxf(mx, v);
    }
    const size_t base = (size_t)B_DIM * T_DIM * O_DIM;
    out[base]     = 0.5f * s / (float)((size_t)B_DIM * T_DIM * O_DIM); // loss
    out[base + 1] = mx / (float)(B_DIM * O_DIM);                       // spread loss
  }
}

// ============================== launcher ====================================
extern "C" void kernel_launch(void* const* d_in, const int* in_sizes, int n_in,
                              void* d_out, int out_size, void* d_ws, size_t ws_size,
                              hipStream_t stream) {
  const float* x    = (const float*)d_in[0];
  const float* w    = (const float*)d_in[1];
  const float* v    = (const float*)d_in[2];
  const float* beta = (const float*)d_in[3];
  const float* bvec = (const float*)d_in[4];
  // d_in[5] = sigma : only used by the backward surrogate, not the forward outputs
  float* out = (float*)d_out;

  // workspace layout (floats)
  float* E        = (float*)d_ws;                        // 1024*1024
  float* mem      = E    + (size_t)O_DIM * O_DIM;        // 128*1024
  float* spkA     = mem  + (size_t)B_DIM * O_DIM;        // 128*1024
  float* spkB     = spkA + (size_t)B_DIM * O_DIM;        // 128*1024
  float* tsum     = spkB + (size_t)B_DIM * O_DIM;        // 128 (T<=128)
  float* inv_norm = tsum + 128;                          // 1024

  // zero mem, spkA, spkB, tsum (contiguous)
  const int nz = B_DIM * O_DIM * 3 + 128;
  init_zero_kernel<<<(nz + 255) / 256, 256, 0, stream>>>(mem, nz);

  // h = x @ w  -> written into d_out (later overwritten in place by spikes)
  gemm_h_kernel<<<dim3(O_DIM / 64, (B_DIM * T_DIM) / 128), 128, 0, stream>>>(x, w, out);

  // E = (1-beta)*v - beta*(w^T w), inv_norm
  gemm_e_kernel<<<dim3(O_DIM / 64, O_DIM / 128), 128, 0, stream>>>(w, v, beta, E, inv_norm);

  // sequential scan: one launch per time step (grid-wide dependency)
  for (int t = 0; t < T_DIM; ++t) {
    const float* sin = (t & 1) ? spkB : spkA;
    float*       so  = (t & 1) ? spkA : spkB;
    scan_step_kernel<<<dim3(O_DIM / 64, B_DIM / 64), 128, 0, stream>>>(
        sin, E, beta, bvec, inv_norm, mem, so, out, tsum, t);
  }

  finalize_kernel<<<1, 32, 0, stream>>>(tsum, out);
}